// QuasiPolyLayer_26551487824340
// MI455X (gfx1250) — compile-verified
//
#include <hip/hip_runtime.h>

// QuasiPolyLayer, B=IN=OUT=512, T=3.
// out[b,o] = bias[o] + Σ_i tw0·x^p  +  Σ_i (p>=3)·tw1·x  +  Σ_i (p>=4)·tw2·x²
// Pow term: 134M v_exp_f32 (trans-bound). GEMM terms: v_wmma_f32_16x16x4_f32.
// All data L2-resident (~6MB << 192MB); HBM is a non-factor.

#define IN_DIM   512
#define OUT_DIM  512
#define LDS_PITCH 516   // 512 + 4-float pad: rows land on distinct banks (stride-4 banks)

typedef __attribute__((ext_vector_type(2))) float v2f;
typedef __attribute__((ext_vector_type(4))) float v4f;
typedef __attribute__((ext_vector_type(8))) float v8f;

__device__ __forceinline__ float poly_mask(float p, float thresh) {
    // reference: n_extra = min(floor(p)-1, 3); mask = (thresh < n_extra)
    float n = fminf(__builtin_floorf(p) - 1.0f, 3.0f);
    return (thresh < n) ? 1.0f : 0.0f;
}

__global__ __launch_bounds__(256)
void quasipoly_wmma_kernel(const float* __restrict__ x,      // [512,512]
                           const float* __restrict__ power,  // [512,512]
                           const float* __restrict__ tw,     // [512,512,3]
                           const float* __restrict__ bias,   // [512]
                           float* __restrict__ out)          // [512,512]
{
    __shared__ float xs [16 * LDS_PITCH];   // x rows for A fragments
    __shared__ float lxs[16 * LDS_PITCH];   // log2(x) rows for pow term (8x reuse)

    const int tid    = threadIdx.x;
    const int m0     = blockIdx.x * 16;     // batch-row tile base
    const int nbase  = blockIdx.y * 128;    // out-col group base (8 waves * 16)

    // Cooperative stage of 16 x-rows + their log2 (1M logs total, amortized 128x)
    for (int e = tid; e < 16 * IN_DIM; e += 256) {
        int r = e >> 9;          // /512
        int c = e & (IN_DIM - 1);
        float v = x[(size_t)(m0 + r) * IN_DIM + c];
        xs [r * LDS_PITCH + c] = v;
        lxs[r * LDS_PITCH + c] = __builtin_amdgcn_logf(v);  // v_log_f32 (log2); x=0 -> -inf -> exp2 -> 0
    }
    __syncthreads();

    const int lane  = tid & 31;
    const int wave  = tid >> 5;
    const int half  = lane >> 4;     // 0: K=0,1 / M=0..7   1: K=2,3 / M=8..15
    const int lid   = lane & 15;
    const int rbase = 8 * half;

    const int    ocol = nbase + wave * 16 + lid;           // this lane's N (out column)
    const float* prow = power + (size_t)ocol * IN_DIM;
    const float* trow = tw    + (size_t)ocol * IN_DIM * 3;

    v8f acc = {};   // C/D tile: lane N=lid, VGPR r -> M = rbase + r

    for (int k0 = 0; k0 < IN_DIM; k0 += 4) {
        // ---------------- WMMA fragments (f32 16x16x4) ----------------
        const int kh = k0 + 2 * half;
        v2f a1; a1.x = xs[lid * LDS_PITCH + kh];
                a1.y = xs[lid * LDS_PITCH + kh + 1];
        v2f a2 = a1 * a1;                                   // X² fragment in-register

        // power / term_weights for this lane's output row, k0..k0+3
        v4f p4 = *(const v4f*)(prow + k0);                  // 16B aligned
        v4f t0 = *(const v4f*)(trow + (size_t)k0 * 3);      // tw[(o,k0..k0+3), t] flat, 48B
        v4f t1 = *(const v4f*)(trow + (size_t)k0 * 3 + 4);
        v4f t2 = *(const v4f*)(trow + (size_t)k0 * 3 + 8);
        // flat index 3j+t : j=k-k0, t=term. Select this half's K-pair (j = 2*half, 2*half+1).
        float pB0 = half ? p4.z : p4.x;
        float pB1 = half ? p4.w : p4.y;
        float w1a = half ? t1.w : t0.y;   // tw1 @ j=2h
        float w1b = half ? t2.z : t1.x;   // tw1 @ j=2h+1
        float w2a = half ? t2.x : t0.z;   // tw2 @ j=2h
        float w2b = half ? t2.w : t1.y;   // tw2 @ j=2h+1

        v2f b1; b1.x = poly_mask(pB0, 1.0f) * w1a;
                b1.y = poly_mask(pB1, 1.0f) * w1b;
        v2f b2; b2.x = poly_mask(pB0, 2.0f) * w2a;
                b2.y = poly_mask(pB1, 2.0f) * w2b;

        // D = A*B + C on the matrix pipe (args: neg_a, A, neg_b, B, c_mod, C, reuse_a, reuse_b)
        acc = __builtin_amdgcn_wmma_f32_16x16x4_f32(false, a1, false, b1, (short)0, acc, false, false);
        acc = __builtin_amdgcn_wmma_f32_16x16x4_f32(false, a2, false, b2, (short)0, acc, false, false);

        // ---------------- pow term: acc[r] += Σ_j tw0·2^(p·log2 x) ----------------
        float w00 = t0.x, w01 = t0.w, w02 = t1.z, w03 = t2.y;  // tw0 @ j=0..3
        #pragma unroll
        for (int r = 0; r < 8; ++r) {
            // lanes 0-15 read identical addresses (broadcast), likewise 16-31; 16B-aligned b128
            v4f lx = *(const v4f*)&lxs[(rbase + r) * LDS_PITCH + k0];
            float s;
            s  = w00 * __builtin_amdgcn_exp2f(p4.x * lx.x);
            s += w01 * __builtin_amdgcn_exp2f(p4.y * lx.y);
            s += w02 * __builtin_amdgcn_exp2f(p4.z * lx.z);
            s += w03 * __builtin_amdgcn_exp2f(p4.w * lx.w);
            acc[r] += s;
        }
    }

    const float bo = bias[ocol];
    #pragma unroll
    for (int r = 0; r < 8; ++r) {
        int m = m0 + rbase + r;
        out[(size_t)m * OUT_DIM + ocol] = acc[r] + bo;   // 16 lanes -> 64B coalesced per row
    }
}

extern "C" void kernel_launch(void* const* d_in, const int* in_sizes, int n_in,
                              void* d_out, int out_size, void* d_ws, size_t ws_size,
                              hipStream_t stream) {
    (void)in_sizes; (void)n_in; (void)out_size; (void)d_ws; (void)ws_size;
    const float* x     = (const float*)d_in[0];  // (512,512)
    const float* power = (const float*)d_in[1];  // (512,512)
    const float* tw    = (const float*)d_in[2];  // (512,512,3)
    // d_in[3] = bias_before: unused by the reference computation
    const float* bias  = (const float*)d_in[4];  // (512,)
    float* out = (float*)d_out;

    dim3 grid(512 / 16, 512 / 128);   // 32 batch tiles x 4 column groups (8 waves * 16 cols)
    quasipoly_wmma_kernel<<<grid, 256, 0, stream>>>(x, power, tw, bias, out);
}